// MultiCBR_13675175871016
// MI455X (gfx1250) — compile-verified
//
#include <hip/hip_runtime.h>
#include <math.h>

// ---------------------------------------------------------------------------
// MultiCBR inference for MI455X (gfx1250, wave32).
// Memory-bound COO SpMM pipeline; WMMA f32 16x16x4 used to compute row Gram
// diagonals (sum of squares) for the L2 normalization; global_prefetch_b8
// used to hide gather latency; native f32 global atomics for segment sums.
// ---------------------------------------------------------------------------

#define NU 30000
#define NB 10000
#define NI 60000
#define DD 64

typedef __attribute__((ext_vector_type(2))) float v2f;
typedef __attribute__((ext_vector_type(8))) float v8f;

// ------------------------------ elementwise --------------------------------

__global__ void fill0_kernel(float* __restrict__ p, long n) {
    long i = ((long)blockIdx.x * blockDim.x + threadIdx.x) * 4;
    if (i < n) *(float4*)(p + i) = make_float4(0.f, 0.f, 0.f, 0.f);
}

// dst = c * src   (c read from device mem; cp==nullptr -> 1.0)
__global__ void scale_copy_kernel(float* __restrict__ dst, const float* __restrict__ src,
                                  const float* __restrict__ cp, int cidx, long n) {
    long i = ((long)blockIdx.x * blockDim.x + threadIdx.x) * 4;
    if (i >= n) return;
    float c = cp ? cp[cidx] : 1.0f;
    float4 v = *(const float4*)(src + i);
    v.x *= c; v.y *= c; v.z *= c; v.w *= c;
    *(float4*)(dst + i) = v;
}

// out = mc[ia]*x + mc[ib]*y
__global__ void combine2_kernel(float* __restrict__ out,
                                const float* __restrict__ x, const float* __restrict__ y,
                                const float* __restrict__ mc, int ia, int ib, long n) {
    long i = ((long)blockIdx.x * blockDim.x + threadIdx.x) * 4;
    if (i >= n) return;
    float a = mc[ia], b = mc[ib];
    float4 xv = *(const float4*)(x + i);
    float4 yv = *(const float4*)(y + i);
    float4 o;
    o.x = a * xv.x + b * yv.x;
    o.y = a * xv.y + b * yv.y;
    o.z = a * xv.z + b * yv.z;
    o.w = a * xv.w + b * yv.w;
    *(float4*)(out + i) = o;
}

// ------------------------------ sparse SpMM --------------------------------

__device__ __forceinline__ void atomAddF(float* p, float v) {
#if defined(__HIP_PLATFORM_AMD__) || defined(__HIP__)
    // guaranteed native global_atomic_add_f32 (no-fine-grained metadata)
    unsafeAtomicAdd(p, v);
#else
    __hip_atomic_fetch_add(p, v, __ATOMIC_RELAXED, __HIP_MEMORY_SCOPE_AGENT);
#endif
}

// out[rows[e]] += scale * vals[e] * feat[cols[e]], D=64.
// 16 threads per edge, float4 per thread -> 256B-coalesced gather segments.
__global__ void spmm_atomic_kernel(const int* __restrict__ rows, const int* __restrict__ cols,
                                   const float* __restrict__ vals,
                                   const float* __restrict__ feat, float* __restrict__ out,
                                   int ne, const float* __restrict__ sp, int sidx) {
    long t  = (long)blockIdx.x * blockDim.x + threadIdx.x;
    int  e  = (int)(t >> 4);
    int  c4 = ((int)t & 15) * 4;
    if (e >= ne) return;

    // prefetch a future edge's gather target into cache (global_prefetch_b8)
    int ep = e + 1024;
    if (ep < ne) {
        int cp = cols[ep];
        __builtin_prefetch(feat + (size_t)cp * DD + c4, 0, 0);
    }

    int   r = rows[e];
    int   c = cols[e];
    float v = vals[e] * (sp ? sp[sidx] : 1.0f);

    float4 f = *(const float4*)(feat + (size_t)c * DD + c4);
    float* o = out + (size_t)r * DD + c4;
    atomAddF(o + 0, v * f.x);
    atomAddF(o + 1, v * f.y);
    atomAddF(o + 2, v * f.z);
    atomAddF(o + 3, v * f.w);
}

// --------------------- L2-normalize + accumulate (WMMA) --------------------
// acc[r] += coef * x[r] / max(||x[r]||_2, 1e-12), one 16-row tile per wave.
// Row sums-of-squares via diag(X Xt): V_WMMA_F32_16X16X4_F32 with the SAME
// register pair fed as A and B (the 32-bit A and B VGPR layouts share the
// (vgpr, lane-half) -> K map, lane&15 -> M resp. N), accumulated over 16
// K-chunks of the 16x64 tile.

__global__ void l2norm_acc_kernel(float* __restrict__ acc, const float* __restrict__ x,
                                  const float* __restrict__ cp, int cidx, int ntiles) {
    int lane = threadIdx.x & 31;
    int wave = blockIdx.x * (blockDim.x >> 5) + (threadIdx.x >> 5);
    if (wave >= ntiles) return;   // wave-uniform: EXEC stays all-ones for WMMA

    const float* xt  = x + (size_t)wave * (16 * DD);
    int row = lane & 15;          // M (and N) index for this lane
    int kof = (lane >> 4) << 1;   // lane-half -> K offset {0,2}
    const float* ap = xt + row * DD + kof;

    float s;  // row sum of squares for 'row'
#if __has_builtin(__builtin_amdgcn_wmma_f32_16x16x4_f32)
    v8f cacc = {0.f, 0.f, 0.f, 0.f, 0.f, 0.f, 0.f, 0.f};
#pragma unroll
    for (int j = 0; j < 16; ++j) {
        v2f a = *(const v2f*)(ap + 4 * j);   // A[m][4j+kof], A[m][4j+kof+1]
        cacc = __builtin_amdgcn_wmma_f32_16x16x4_f32(
            /*neg_a=*/false, a, /*neg_b=*/false, a,
            /*c_mod=*/(short)0, cacc, /*reuse_a=*/false, /*reuse_b=*/false);
    }
    // diag(C): M<8 at (vgpr M, lane M); M>=8 at (vgpr M-8, lane M+16)
    int di = lane & 7;
    float d = cacc[0];
    d = (di == 1) ? cacc[1] : d;
    d = (di == 2) ? cacc[2] : d;
    d = (di == 3) ? cacc[3] : d;
    d = (di == 4) ? cacc[4] : d;
    d = (di == 5) ? cacc[5] : d;
    d = (di == 6) ? cacc[6] : d;
    d = (di == 7) ? cacc[7] : d;
    int src = (row < 8) ? row : (row + 16);
    s = __shfl(d, src, 32);
#else
    // fallback: lane-pair shuffle reduction
    float partial = 0.f;
    const float* xr0 = xt + row * DD + (lane >> 4) * 32;
    for (int q = 0; q < 32; ++q) { float vv = xr0[q]; partial += vv * vv; }
    s = partial + __shfl_xor(partial, 16, 32);
#endif

    float coef = cp ? cp[cidx] : 1.0f;
    float k = coef / fmaxf(sqrtf(s), 1e-12f);

    int cb = (lane >> 4) * 32;    // this lane writes cols [cb, cb+32) of 'row'
    const float* xr = xt + row * DD + cb;
    float* ar = acc + (size_t)wave * (16 * DD) + row * DD + cb;
#pragma unroll
    for (int q = 0; q < 8; ++q) {
        float4 xv = *(const float4*)(xr + 4 * q);
        float4 av = *(float4*)(ar + 4 * q);
        av.x += k * xv.x; av.y += k * xv.y; av.z += k * xv.z; av.w += k * xv.w;
        *(float4*)(ar + 4 * q) = av;
    }
}

// ------------------------------- launcher ----------------------------------

static inline unsigned cdiv(long a, long b) { return (unsigned)((a + b - 1) / b); }

extern "C" void kernel_launch(void* const* d_in, const int* in_sizes, int n_in,
                              void* d_out, int out_size, void* d_ws, size_t ws_size,
                              hipStream_t stream) {
    const float* users_f  = (const float*)d_in[0];
    const float* bundles_f= (const float*)d_in[1];
    const float* items_f  = (const float*)d_in[2];
    const int*   ub_r = (const int*)d_in[3];
    const int*   ub_c = (const int*)d_in[4];
    const float* ub_v = (const float*)d_in[5];
    const int*   ui_r = (const int*)d_in[6];
    const int*   ui_c = (const int*)d_in[7];
    const float* ui_v = (const float*)d_in[8];
    const int*   bi_r = (const int*)d_in[9];
    const int*   bi_c = (const int*)d_in[10];
    const float* bi_v = (const float*)d_in[11];
    const int*   biag_r = (const int*)d_in[12];
    const int*   biag_c = (const int*)d_in[13];
    const float* biag_v = (const float*)d_in[14];
    const int*   uiag_r = (const int*)d_in[15];
    const int*   uiag_c = (const int*)d_in[16];
    const float* uiag_v = (const float*)d_in[17];
    const float* ub_coef = (const float*)d_in[18];
    const float* ui_coef = (const float*)d_in[19];
    const float* bi_coef = (const float*)d_in[20];
    const float* modal   = (const float*)d_in[21];

    const int ne_ub   = in_sizes[3];
    const int ne_ui   = in_sizes[6];
    const int ne_bi   = in_sizes[9];
    const int ne_biag = in_sizes[12];
    const int ne_uiag = in_sizes[15];

    // workspace layout (floats): 3 persistent accumulators + 2 ping-pong bufs
    float* ws = (float*)d_ws;
    float* ub_acc = ws;                              // (NU+NB) x 64
    float* ui_acc = ub_acc + (long)(NU + NB) * DD;   // (NU+NI) x 64
    float* bi_acc = ui_acc + (long)(NU + NI) * DD;   // (NB+NI) x 64
    float* F0     = bi_acc + (long)(NB + NI) * DD;   // (NU+NI) x 64 max
    float* F1     = F0     + (long)(NU + NI) * DD;   // (NU+NI) x 64 max

    const int TPB = 256;

    auto propagate = [&](const int* rows, const int* cols, const float* vals, int ne,
                         const float* a, int na, const float* b, int nb,
                         const float* coefs, float* acc) {
        long n64  = (long)(na + nb) * DD;
        long na64 = (long)na * DD;
        long nb64 = (long)nb * DD;
        int  ntiles = (na + nb) / 16;
        // feat0 = concat(a, b)
        scale_copy_kernel<<<cdiv(na64, TPB * 4L), TPB, 0, stream>>>(F0, a, nullptr, 0, na64);
        scale_copy_kernel<<<cdiv(nb64, TPB * 4L), TPB, 0, stream>>>(F0 + na64, b, nullptr, 0, nb64);
        // layer-0 term (un-normalized)
        scale_copy_kernel<<<cdiv(n64, TPB * 4L), TPB, 0, stream>>>(acc, F0, coefs, 0, n64);
        // layer 1
        fill0_kernel<<<cdiv(n64, TPB * 4L), TPB, 0, stream>>>(F1, n64);
        spmm_atomic_kernel<<<cdiv((long)ne * 16, TPB), TPB, 0, stream>>>(
            rows, cols, vals, F0, F1, ne, nullptr, 0);
        l2norm_acc_kernel<<<cdiv(ntiles, TPB / 32L), TPB, 0, stream>>>(acc, F1, coefs, 1, ntiles);
        // layer 2
        fill0_kernel<<<cdiv(n64, TPB * 4L), TPB, 0, stream>>>(F0, n64);
        spmm_atomic_kernel<<<cdiv((long)ne * 16, TPB), TPB, 0, stream>>>(
            rows, cols, vals, F1, F0, ne, nullptr, 0);
        l2norm_acc_kernel<<<cdiv(ntiles, TPB / 32L), TPB, 0, stream>>>(acc, F0, coefs, 2, ntiles);
    };

    propagate(ub_r, ub_c, ub_v, ne_ub, users_f,   NU, bundles_f, NB, ub_coef, ub_acc);
    propagate(ui_r, ui_c, ui_v, ne_ui, users_f,   NU, items_f,   NI, ui_coef, ui_acc);
    propagate(bi_r, bi_c, bi_v, ne_bi, bundles_f, NB, items_f,   NI, bi_coef, bi_acc);

    float* out_u = (float*)d_out;                 // NU x 64
    float* out_b = out_u + (long)NU * DD;         // NB x 64

    // users_rep = m0*ub_u + m1*ui_u + m2*(ui_agg @ bi_i)
    combine2_kernel<<<cdiv((long)NU * DD, TPB * 4L), TPB, 0, stream>>>(
        out_u, ub_acc, ui_acc, modal, 0, 1, (long)NU * DD);
    spmm_atomic_kernel<<<cdiv((long)ne_uiag * 16, TPB), TPB, 0, stream>>>(
        uiag_r, uiag_c, uiag_v, bi_acc + (long)NB * DD, out_u, ne_uiag, modal, 2);

    // bundles_rep = m0*ub_b + m2*bi_b + m1*(bi_agg @ ui_i)
    combine2_kernel<<<cdiv((long)NB * DD, TPB * 4L), TPB, 0, stream>>>(
        out_b, ub_acc + (long)NU * DD, bi_acc, modal, 0, 2, (long)NB * DD);
    spmm_atomic_kernel<<<cdiv((long)ne_biag * 16, TPB), TPB, 0, stream>>>(
        biag_r, biag_c, biag_v, ui_acc + (long)NU * DD, out_b, ne_biag, modal, 1);
}